// HybridGNNModel_1142461301283
// MI455X (gfx1250) — compile-verified
//
#include <hip/hip_runtime.h>
#include <hip/hip_bf16.h>

typedef __attribute__((ext_vector_type(2))) float v2f;
typedef __attribute__((ext_vector_type(8))) float v8f;

// ---------------- small helper kernels ----------------

__global__ void k_zero(float* __restrict__ p, int n) {
    int i = blockIdx.x * blockDim.x + threadIdx.x;
    if (i < n) p[i] = 0.0f;
}

// deg[n] = #edges with dst==n, plus 1 self-loop per node
__global__ void k_degree(const int* __restrict__ dst, int E, int N,
                         float* __restrict__ deg) {
    int i = blockIdx.x * blockDim.x + threadIdx.x;
    if (i < E) {
        atomicAdd(&deg[dst[i]], 1.0f);
    } else if (i < E + N) {
        atomicAdd(&deg[i - E], 1.0f);
    }
}

__global__ void k_dinv(const float* __restrict__ deg, float* __restrict__ dinv, int N) {
    int i = blockIdx.x * blockDim.x + threadIdx.x;
    if (i < N) {
        float d = deg[i];
        dinv[i] = (d > 0.0f) ? rsqrtf(d) : 0.0f;
    }
}

// w_src[n] = sum over edges e with src_e==n of dinv[src]*dinv[dst]  (+ self-loop dinv[n]^2)
__global__ void k_wsrc(const int* __restrict__ src, const int* __restrict__ dst,
                       int E, int N, const float* __restrict__ dinv,
                       float* __restrict__ wsrc) {
    int i = blockIdx.x * blockDim.x + threadIdx.x;
    if (i < E) {
        int s = src[i], d = dst[i];
        atomicAdd(&wsrc[s], dinv[s] * dinv[d]);
    } else if (i < E + N) {
        int n = i - E;
        atomicAdd(&wsrc[n], dinv[n] * dinv[n]);
    }
}

// ---------------- fused GEMM + weighted node reduction (WMMA fp32) ----------------
// s[b,:] = sum_n w_src[n] * relu(x[b,n,:] @ W_feat + b_feat)
// One wave per 16-node tile. H = D = 128.
__global__ __launch_bounds__(256)
void k_gemm_reduce(const float* __restrict__ x, const float* __restrict__ Wf,
                   const float* __restrict__ bf, const float* __restrict__ wsrc,
                   float* __restrict__ svec, int N, int ntiles) {
    __shared__ float Wlds[128 * 128];  // 64 KB: full W_feat, row-major
    int tid = threadIdx.x;
    for (int i = tid; i < 128 * 128; i += 256) Wlds[i] = Wf[i];
    __syncthreads();

    int wave = tid >> 5;
    int lane = tid & 31;
    int t = blockIdx.x * 8 + wave;          // tile id, uniform per wave
    int tiles_total = 2 * ntiles;
    if (t >= tiles_total) return;

    int b  = t / ntiles;
    int n0 = (t - b * ntiles) * 16;
    int half = lane >> 4;                   // 0: K+0/K+1 ; 1: K+2/K+3 (ISA 16x4 f32 layout)
    int lcol = lane & 15;

    int row = n0 + lcol;
    if (row >= N) row = N - 1;              // clamp (weights below are zeroed)

    // A fragments: lane holds x[row][4s + 2*half + {0,1}] for s = 0..31 (K = 128)
    const float* xrow = x + ((size_t)b * N + row) * 128 + 2 * half;
    v2f a[32];
#pragma unroll
    for (int s = 0; s < 32; ++s) {
        a[s] = *(const v2f*)(xrow + 4 * s);
    }

    // row weights: C/D layout has lane-half covering rows r+8*half
    float wrow[8];
#pragma unroll
    for (int r = 0; r < 8; ++r) {
        int rr = n0 + r + 8 * half;
        wrow[r] = (rr < N) ? wsrc[rr] : 0.0f;
    }

#pragma unroll 1
    for (int j = 0; j < 8; ++j) {           // 8 column tiles of 16 (H = 128)
        int colbase = j * 16 + lcol;
        v8f acc = {0.f, 0.f, 0.f, 0.f, 0.f, 0.f, 0.f, 0.f};
#pragma unroll
        for (int s = 0; s < 32; ++s) {
            int k = 4 * s + 2 * half;
            v2f bb;
            bb[0] = Wlds[k * 128 + colbase];        // B[k  ][col]
            bb[1] = Wlds[(k + 1) * 128 + colbase];  // B[k+1][col]
            acc = __builtin_amdgcn_wmma_f32_16x16x4_f32(
                false, a[s], false, bb, (short)0, acc, false, false);
        }
        float bias = bf[colbase];
        float part = 0.0f;
#pragma unroll
        for (int r = 0; r < 8; ++r) {
            float h = acc[r] + bias;
            h = (h > 0.0f) ? h : 0.0f;      // relu
            part += wrow[r] * h;            // weighted row reduction
        }
        part += __shfl_xor(part, 16, 32);   // combine lane halves (same col, rows 0-7 vs 8-15)
        if (lane < 16) atomicAdd(&svec[b * 128 + colbase], part);
    }
}

// ---------------- tiny head: pooled -> fc1 -> fc2 ----------------
__global__ void k_head(const float* __restrict__ svec,
                       const float* __restrict__ Wg, const float* __restrict__ bg,
                       const float* __restrict__ W1, const float* __restrict__ b1,
                       const float* __restrict__ W2, const float* __restrict__ b2,
                       float* __restrict__ out, int N) {
    __shared__ float pooled[256];
    __shared__ float z[256];
    int tid = threadIdx.x;                  // 256 threads: b = tid/128, c = tid%128
    int b = tid >> 7, c = tid & 127;
    float invN = 1.0f / (float)N;

    float acc = 0.0f;
    for (int k = 0; k < 128; ++k) acc += svec[b * 128 + k] * Wg[k * 128 + c];
    pooled[tid] = acc * invN + bg[c];
    __syncthreads();

    float acc2 = 0.0f;
    for (int k = 0; k < 128; ++k) acc2 += pooled[b * 128 + k] * W1[k * 128 + c];
    acc2 += b1[c];
    z[tid] = (acc2 > 0.0f) ? acc2 : 0.0f;
    __syncthreads();

    if (tid < 2) {
        float l = 0.0f;
        for (int k = 0; k < 128; ++k) l += z[tid * 128 + k] * W2[k];
        out[tid] = l + b2[0];
    }
}

// ---------------- launch ----------------
extern "C" void kernel_launch(void* const* d_in, const int* in_sizes, int n_in,
                              void* d_out, int out_size, void* d_ws, size_t ws_size,
                              hipStream_t stream) {
    const float* x  = (const float*)d_in[0];
    const int*   ei = (const int*)  d_in[1];
    const float* Wf = (const float*)d_in[2];
    const float* bf = (const float*)d_in[3];
    const float* Wg = (const float*)d_in[4];
    const float* bg = (const float*)d_in[5];
    const float* W1 = (const float*)d_in[6];
    const float* b1 = (const float*)d_in[7];
    const float* W2 = (const float*)d_in[8];
    const float* b2 = (const float*)d_in[9];

    int E = in_sizes[1] / 2;        // edge_index is [2, E]
    int N = in_sizes[0] / 256;      // x is [2, N, 128]
    const int* src = ei;
    const int* dst = ei + E;

    float* deg  = (float*)d_ws;     // [N]
    float* dinv = deg  + N;         // [N]
    float* wsrc = dinv + N;         // [N]
    float* svec = wsrc + N;         // [2*128]

    int zn = 3 * N + 256;           // zero deg/dinv/wsrc/svec every call (determinism)
    k_zero<<<(zn + 255) / 256, 256, 0, stream>>>(deg, zn);

    int m = E + N;
    k_degree<<<(m + 255) / 256, 256, 0, stream>>>(dst, E, N, deg);
    k_dinv  <<<(N + 255) / 256, 256, 0, stream>>>(deg, dinv, N);
    k_wsrc  <<<(m + 255) / 256, 256, 0, stream>>>(src, dst, E, N, dinv, wsrc);

    int ntiles = (N + 15) / 16;
    int tiles_total = 2 * ntiles;
    k_gemm_reduce<<<(tiles_total + 7) / 8, 256, 0, stream>>>(x, Wf, bf, wsrc, svec, N, ntiles);

    k_head<<<1, 256, 0, stream>>>(svec, Wg, bg, W1, b1, W2, b2, (float*)d_out, N);
}